// RestrictedHighOrderActivationA_85220741087982
// MI455X (gfx1250) — compile-verified
//
#include <hip/hip_runtime.h>

// ---------------------------------------------------------------------------
// Simplex-interpolation activation as per-group 16x16 WMMA GEMMs (gfx1250).
//
// out[b, g*16 + d] = sum_k coef[b,g,k] * params[g, idx[b,g,k], d]
//   == (scatter coef into sparse row W[b, 0..15]) @ params[g]  (16x16 f32 GEMM)
//
// Memory-bound problem (~84 MB traffic, ~3.6 us at 23.3 TB/s); f32 throughout.
// Each wave: one group g, B-operand (params[g]) held in 8 VGPRs and reused
// across 4 batch tiles; 4x V_WMMA_F32_16X16X4_F32 accumulate K=16 per tile.
// ---------------------------------------------------------------------------

typedef __attribute__((ext_vector_type(2))) float v2f;
typedef __attribute__((ext_vector_type(8))) float v8f;

namespace {
constexpr int G       = 256;
constexpr int ARITY   = 4;
constexpr int D       = 16;
constexpr int BATCH   = 4096;
constexpr int XW      = G * ARITY;   // 1024 floats per X row
constexpr int OUTW    = G * D;       // 4096 floats per out row
constexpr int WAVES_PER_BLOCK = 8;   // 256 threads
constexpr int TILES_PER_WAVE  = 4;   // 16-row batch tiles per wave
constexpr int BLOCKS_Y = BATCH / 16 / (WAVES_PER_BLOCK * TILES_PER_WAVE); // 8
}

__global__ __launch_bounds__(256)
void simplex_wmma_kernel(const float* __restrict__ X,
                         const float* __restrict__ params,
                         float* __restrict__ out)
{
    const int lane = threadIdx.x & 31;
    const int wave = threadIdx.x >> 5;
    const int g    = blockIdx.x;
    const int half = lane >> 4;      // 0: lanes 0-15 (K=0,1), 1: lanes 16-31 (K=2,3)
    const int m    = lane & 15;      // row (A) / column (B,C) within the 16x16 tile

    // ---- B operand: params[g] (16x16, K=p rows, N=d cols) as four 4x16 chunks.
    // 32-bit B 4x16 layout: VGPR0 = K=(chunk*4 + half*2), VGPR1 = K+1; N = lane&15.
    const float* __restrict__ P = params + (size_t)g * (16 * D);
    v2f bmat[4];
#pragma unroll
    for (int kk = 0; kk < 4; ++kk) {
        const int p0 = kk * 4 + half * 2;
        bmat[kk].x = P[p0 * D + m];
        bmat[kk].y = P[(p0 + 1) * D + m];
    }

#pragma unroll
    for (int t = 0; t < TILES_PER_WAVE; ++t) {
        const int tile = (blockIdx.y * WAVES_PER_BLOCK + wave) * TILES_PER_WAVE + t;
        const int b0   = tile * 16;

        // ---- Per-row prep (both half-waves redundantly prep row m): load 4
        // values, sort-4 network (stable-enough: swap only on strict >),
        // coefficients = [min, diffs], corner indices = reverse-OR of 2^argsort.
        const float4 xr = *reinterpret_cast<const float4*>(
            X + (size_t)(b0 + m) * XW + (size_t)g * ARITY);
        float v0 = xr.x, v1 = xr.y, v2 = xr.z, v3 = xr.w;
        int   j0 = 0, j1 = 1, j2 = 2, j3 = 3;
        if (v0 > v1) { float tv=v0; v0=v1; v1=tv; int tj=j0; j0=j1; j1=tj; }
        if (v2 > v3) { float tv=v2; v2=v3; v3=tv; int tj=j2; j2=j3; j3=tj; }
        if (v0 > v2) { float tv=v0; v0=v2; v2=tv; int tj=j0; j0=j2; j2=tj; }
        if (v1 > v3) { float tv=v1; v1=v3; v3=tv; int tj=j1; j1=j3; j3=tj; }
        if (v1 > v2) { float tv=v1; v1=v2; v2=tv; int tj=j1; j1=j2; j2=tj; }

        const float c0 = v0, c1 = v1 - v0, c2 = v2 - v1, c3 = v3 - v2;
        const int   i3 = 1 << j3;
        const int   i2 = i3 | (1 << j2);
        const int   i1 = i2 | (1 << j1);
        const int   i0 = 15;                    // i1 | (1 << j0) == full mask

        // Scatter: W[row m, p] — corner indices are distinct, so sum-of-selects.
        auto wsel = [&](int p) -> float {
            float r = (i0 == p) ? c0 : 0.0f;
            r += (i1 == p) ? c1 : 0.0f;
            r += (i2 == p) ? c2 : 0.0f;
            r += (i3 == p) ? c3 : 0.0f;
            return r;
        };

        // ---- 16x16x16 f32 GEMM as four V_WMMA_F32_16X16X4_F32.
        // 32-bit A 16x4 layout: VGPR0 = K=(chunk*4 + half*2), VGPR1 = K+1; M = lane&15.
        v8f acc = {};
#pragma unroll
        for (int kk = 0; kk < 4; ++kk) {
            const int p0 = kk * 4 + half * 2;
            v2f a;
            a.x = wsel(p0);
            a.y = wsel(p0 + 1);
            // (neg_a, A, neg_b, B, c_mod, C, reuse_a, reuse_b)
            acc = __builtin_amdgcn_wmma_f32_16x16x4_f32(
                false, a, false, bmat[kk], (short)0, acc, false, false);
        }

        // ---- Store C tile. 16x16 f32 C layout: VGPR j -> M = j + half*8, N = m.
        float* __restrict__ orow =
            out + (size_t)(b0 + half * 8) * OUTW + (size_t)g * D + m;
#pragma unroll
        for (int j = 0; j < 8; ++j) {
            orow[(size_t)j * OUTW] = acc[j];
        }
    }
}

extern "C" void kernel_launch(void* const* d_in, const int* in_sizes, int n_in,
                              void* d_out, int out_size, void* d_ws, size_t ws_size,
                              hipStream_t stream) {
    (void)in_sizes; (void)n_in; (void)out_size; (void)d_ws; (void)ws_size;
    const float* X      = (const float*)d_in[0];
    const float* params = (const float*)d_in[1];
    float*       out    = (float*)d_out;

    dim3 grid(G, BLOCKS_Y);            // 256 groups x 8 batch-slices
    dim3 block(32 * WAVES_PER_BLOCK);  // 8 waves
    simplex_wmma_kernel<<<grid, block, 0, stream>>>(X, params, out);
}